// GCNLayer_27642409517682
// MI455X (gfx1250) — compile-verified
//
#include <hip/hip_runtime.h>

typedef __attribute__((ext_vector_type(2))) float v2f;
typedef __attribute__((ext_vector_type(8))) float v8f;

#define D 128
#define LDS_STRIDE 132   // 128 + 4 dword pad: rows shift 4 banks -> conflict-free b64 reads

// ---------------- zero h ----------------
__global__ void gcn_zero_kernel(float4* __restrict__ p, long long n4) {
    long long i = (long long)blockIdx.x * blockDim.x + threadIdx.x;
    if (i < n4) p[i] = make_float4(0.f, 0.f, 0.f, 0.f);
}

// ---------------- scatter-add: h[dst] += feature[src] ----------------
// one wave32 per edge; lane handles 4 consecutive floats (float4 gather + 4 atomics)
__global__ void __launch_bounds__(256) gcn_scatter_kernel(
    const float* __restrict__ feat,
    const int* __restrict__ src,
    const int* __restrict__ dst,
    float* __restrict__ h,
    int n_edges) {
    int wave = (int)(((long long)blockIdx.x * blockDim.x + threadIdx.x) >> 5);
    int lane = threadIdx.x & 31;
    if (wave >= n_edges) return;
    int s = src[wave];   // wave-uniform
    int d = dst[wave];   // wave-uniform
    const float4 v = *(const float4*)(feat + (size_t)s * D + lane * 4);
    float* hp = h + (size_t)d * D + lane * 4;
    atomicAdd(hp + 0, v.x);
    atomicAdd(hp + 1, v.y);
    atomicAdd(hp + 2, v.z);
    atomicAdd(hp + 3, v.w);
}

// ---------------- out = relu(h @ W^T + b) via V_WMMA_F32_16X16X4_F32 ----------------
// block: 256 threads = 8 waves; block owns rows [m0, m0+16); wave w owns cols [16w, 16w+16)
__global__ void __launch_bounds__(256) gcn_gemm_bias_relu_kernel(
    const float* __restrict__ h,
    const float* __restrict__ W,   // [D_OUT, D_IN] row-major -> B[k][n] = W[n][k]
    const float* __restrict__ bias,
    float* __restrict__ out) {
    __shared__ float lds[16 * LDS_STRIDE];

    const int m0 = blockIdx.x * 16;
    const int t  = threadIdx.x;

    // stage 16x128 h tile into LDS (512 float4 loads, fully coalesced)
    {
        const float4* hp = (const float4*)(h + (size_t)m0 * D);
        #pragma unroll
        for (int i = t; i < 512; i += 256) {
            int row = i >> 5;        // 0..15
            int c4  = i & 31;        // 0..31 (float4 index)
            float4 v = hp[row * 32 + c4];
            *(float4*)(&lds[row * LDS_STRIDE + c4 * 4]) = v;  // row stride 528B, 16B aligned
        }
    }
    __syncthreads();

    const int wv   = t >> 5;
    const int lane = t & 31;
    const int llo  = lane & 15;   // N (and M for A-frag) index within tile
    const int lhi  = lane >> 4;   // selects K pair: 0 -> K{0,1}, 1 -> K{2,3}
    const int n0   = wv * 16;
    const int ncol = n0 + llo;

    // A frag (16x4, M=llo): VGPR0 = A[M][2*lhi], VGPR1 = A[M][2*lhi+1]
    const float* arow = &lds[llo * LDS_STRIDE + 2 * lhi];
    // B frag (4x16, N=llo): VGPR0 = B[2*lhi][N] = W[N][2*lhi], VGPR1 = W[N][2*lhi+1]
    const float* brow = W + (size_t)ncol * D + 2 * lhi;

    v8f c = {};
    #pragma unroll
    for (int k = 0; k < D; k += 4) {
        float2 af = *(const float2*)(arow + k);   // ds_load_b64, conflict-free via pad
        float2 bf = *(const float2*)(brow + k);   // global_load_b64, L2-resident W
        v2f a; a[0] = af.x; a[1] = af.y;
        v2f b; b[0] = bf.x; b[1] = bf.y;
        // 8 args: (neg_a, A, neg_b, B, c_mod, C, reuse_a, reuse_b)
        c = __builtin_amdgcn_wmma_f32_16x16x4_f32(false, a, false, b, (short)0, c,
                                                  false, false);
    }

    const float bv = bias[ncol];
    // D frag: VGPR v -> out[m0 + v + 8*lhi][ncol]
    #pragma unroll
    for (int v = 0; v < 8; ++v) {
        int row = m0 + v + 8 * lhi;
        float val = c[v] + bv;
        out[(size_t)row * D + ncol] = fmaxf(val, 0.0f);
    }
}

extern "C" void kernel_launch(void* const* d_in, const int* in_sizes, int n_in,
                              void* d_out, int out_size, void* d_ws, size_t ws_size,
                              hipStream_t stream) {
    const float* feature = (const float*)d_in[0];
    const int*   src     = (const int*)d_in[1];
    const int*   dst     = (const int*)d_in[2];
    const float* W       = (const float*)d_in[3];
    const float* bias    = (const float*)d_in[4];
    float*       out     = (float*)d_out;
    float*       h       = (float*)d_ws;   // N_NODES * 128 f32 accumulator (51.2 MB)

    const int n_nodes = in_sizes[0] / D;
    const int n_edges = in_sizes[1];

    // 1) h = 0
    long long n4 = (long long)n_nodes * (D / 4);
    gcn_zero_kernel<<<(unsigned)((n4 + 255) / 256), 256, 0, stream>>>((float4*)h, n4);

    // 2) scatter-add (one wave32 per edge -> 8 edges per 256-thread block)
    gcn_scatter_kernel<<<(unsigned)((n_edges + 7) / 8), 256, 0, stream>>>(
        feature, src, dst, h, n_edges);

    // 3) WMMA f32 GEMM + bias + relu (n_nodes = 100000 is divisible by 16)
    gcn_gemm_bias_relu_kernel<<<(unsigned)(n_nodes / 16), 256, 0, stream>>>(
        h, W, bias, out);
}